// MeshSimplificationLoss_11957188952683
// MI455X (gfx1250) — compile-verified
//
#include <hip/hip_runtime.h>
#include <float.h>

// CDNA5 / gfx1250: wave32, WMMA 16x16x4 f32 (A=16x4 in 2 VGPRs, B=4x16 in 2 VGPRs,
// C/D = 16x16 f32 in 8 VGPRs).
typedef __attribute__((ext_vector_type(2))) float v2f;
typedef __attribute__((ext_vector_type(8))) float v8f;

#define THREADS          256
#define WAVES_PER_BLOCK  8      // 256 / wave32
#define ROWS_PER_WAVE    16
#define ROWS_PER_BLOCK   128    // 8 waves * 16 rows
#define COLS_STAGE       1024   // columns staged in LDS per pass (12 KB)

__device__ inline v8f wmma16x16x4_f32(v2f a, v2f b, v8f c) {
  // 8-arg pattern: (neg_a, A, neg_b, B, c_mod, C, reuse_a, reuse_b)
  return __builtin_amdgcn_wmma_f32_16x16x4_f32(
      false, a, false, b, (short)0, c, false, false);
}

// -------------------------------------------------------------------------
// Chamfer one-direction kernel:
//   for each row point r in R (per batch), min over all column points c in C of
//   ||r - c||^2, summed over the wave's 16 rows -> one partial per wave.
// Computes d2 tiles with V_WMMA_F32_16X16X4_F32:
//   A fragment pre-scaled by -2, C accumulator preloaded with |x|^2 + |y|^2,
//   so the WMMA result IS the squared distance tile.
// Requires: nR % 128 == 0, nC % 1024 == 0 (true for N=4096, M=1024).
// -------------------------------------------------------------------------
__global__ __launch_bounds__(THREADS)
void chamfer_min_kernel(const float* __restrict__ R, const float* __restrict__ C,
                        float* __restrict__ partials, int nR, int nC)
{
  __shared__ float sx[COLS_STAGE], sy[COLS_STAGE], sz[COLS_STAGE];

  const int blocksPerBatch = nR / ROWS_PER_BLOCK;
  const int b    = blockIdx.x / blocksPerBatch;
  const int rb   = blockIdx.x % blocksPerBatch;
  const int wave = threadIdx.x >> 5;
  const int lane = threadIdx.x & 31;
  const bool hi  = (lane >= 16);
  const int halfoff = hi ? 8 : 0;
  const int i0   = rb * ROWS_PER_BLOCK + wave * ROWS_PER_WAVE;

  const float* Rb = R + (size_t)b * nR * 3;
  const float* Cb = C + (size_t)b * nC * 3;

  // A fragment (constant for the whole kernel): lane holds row i0+(lane&15),
  // K = {0,1} in lanes 0-15, K = {2,3} in lanes 16-31 (K=3 padded with 0).
  const int ra = i0 + (lane & 15);
  const float px = Rb[ra * 3 + 0];
  const float py = Rb[ra * 3 + 1];
  const float pz = Rb[ra * 3 + 2];
  v2f afrag;
  afrag.x = -2.0f * (hi ? pz : px);            // pre-scale: WMMA yields -2<x,y>
  afrag.y = hi ? 0.0f : (-2.0f * py);

  // |x|^2 for the 8 output rows this lane produces (C layout: rows v / v+8).
  float x2r[8];
#pragma unroll
  for (int v = 0; v < 8; ++v) {
    const int ir = i0 + halfoff + v;
    const float X = Rb[ir * 3 + 0], Y = Rb[ir * 3 + 1], Z = Rb[ir * 3 + 2];
    x2r[v] = X * X + Y * Y + Z * Z;
  }

  float mn[8];
#pragma unroll
  for (int v = 0; v < 8; ++v) mn[v] = FLT_MAX;

  for (int s0 = 0; s0 < nC; s0 += COLS_STAGE) {
    // cooperative stage of 1024 column points into LDS
    for (int t = threadIdx.x; t < COLS_STAGE; t += THREADS) {
      const int j = s0 + t;
      sx[t] = Cb[j * 3 + 0];
      sy[t] = Cb[j * 3 + 1];
      sz[t] = Cb[j * 3 + 2];
    }
    __syncthreads();

    for (int t0 = 0; t0 < COLS_STAGE; t0 += 16) {
      const int cj = t0 + (lane & 15);         // column this lane serves
      const float qx = sx[cj], qy = sy[cj], qz = sz[cj];
      v2f bfrag;                               // B fragment, same half-split as A
      bfrag.x = hi ? qz : qx;
      bfrag.y = hi ? 0.0f : qy;
      const float y2 = qx * qx + qy * qy + qz * qz;

      v8f acc;                                 // C := |x|^2 + |y|^2
#pragma unroll
      for (int v = 0; v < 8; ++v) acc[v] = x2r[v] + y2;

      acc = wmma16x16x4_f32(afrag, bfrag, acc); // acc := d2 tile

#pragma unroll
      for (int v = 0; v < 8; ++v)
        mn[v] = fminf(mn[v], fmaxf(acc[v], 0.0f)); // clamp like jnp.maximum(d2,0)
    }
    __syncthreads();
  }

  // Min-reduce across the 16 lanes sharing each output row
  // (xor masks 1,2,4,8 never cross the 16-lane half boundary).
#pragma unroll
  for (int v = 0; v < 8; ++v) {
#pragma unroll
    for (int off = 1; off < 16; off <<= 1)
      mn[v] = fminf(mn[v], __shfl_xor(mn[v], off, 32));
  }
  float s = 0.0f;
#pragma unroll
  for (int v = 0; v < 8; ++v) s += mn[v];
  s += __shfl_xor(s, 16, 32);                  // rows 0-7 + rows 8-15
  if (lane == 0)
    partials[blockIdx.x * WAVES_PER_BLOCK + wave] = s;  // sum of 16 row-mins
}

// smoothed curvature is identically zero: the reference's einsum builds a 10x10
// Gram matrix of 10 vectors in R^3 (rank <= 3), whose smallest eigenvalue is 0,
// so curv == 0 and the Gaussian smoothing of the zero vector is 0.
__global__ void zero_kernel(float* __restrict__ p, int n) {
  const int i = blockIdx.x * blockDim.x + threadIdx.x;
  if (i < n) p[i] = 0.0f;
}

// Deterministic serial reduction of per-wave partials into total_loss[b]
// (no float atomics -> bitwise identical across graph replays).
__global__ void reduce_kernel(const float* __restrict__ t1p,
                              const float* __restrict__ t2p,
                              float* __restrict__ out,
                              int B, int n1, int n2) {
  const int b = blockIdx.x * blockDim.x + threadIdx.x;
  if (b < B) {
    float s = 0.0f;
    for (int i = 0; i < n1; ++i) s += t1p[b * n1 + i];
    for (int i = 0; i < n2; ++i) s += t2p[b * n2 + i];
    out[b] = s;
  }
}

extern "C" void kernel_launch(void* const* d_in, const int* in_sizes, int n_in,
                              void* d_out, int out_size, void* d_ws, size_t ws_size,
                              hipStream_t stream) {
  const float* p1 = (const float*)d_in[0];   // (B,N,3) f32
  const float* p2 = (const float*)d_in[1];   // (B,M,3) f32
  float* out = (float*)d_out;                // [ total_loss(B) | smoothed(B*N) ]

  const int BN = in_sizes[0] / 3;            // B*N
  const int BM = in_sizes[1] / 3;            // B*M
  const int B  = out_size - BN;              // out_size = B + B*N
  if (B <= 0) return;
  const int N  = BN / B;                     // 4096
  const int M  = BM / B;                     // 1024

  float* t1p = (float*)d_ws;                 // B*N/16 wave partials (p1 -> p2)
  float* t2p = t1p + (size_t)BN / 16;        // B*M/16 wave partials (p2 -> p1)

  // smoothed = 0 (see kernel comment for the rank argument)
  {
    const int n = BN;
    zero_kernel<<<(n + 255) / 256, 256, 0, stream>>>(out + B, n);
  }

  // chamfer term1: rows = points1, cols = points2
  chamfer_min_kernel<<<(B * N) / ROWS_PER_BLOCK, THREADS, 0, stream>>>(
      p1, p2, t1p, N, M);
  // chamfer term2: rows = points2, cols = points1 (distance is symmetric)
  chamfer_min_kernel<<<(B * M) / ROWS_PER_BLOCK, THREADS, 0, stream>>>(
      p2, p1, t2p, M, N);

  reduce_kernel<<<1, 64, 0, stream>>>(t1p, t2p, out, B, N / 16, M / 16);
}